// GhostAttention_12266426597797
// MI455X (gfx1250) — compile-verified
//
#include <hip/hip_runtime.h>
#include <math.h>

#define D_MODEL 1024
#define N_HEAD  16
#define HEAD_DIM 64
#define B_ 4
#define T_ 1024
#define MROWS (B_*T_)   /* 4096 */

typedef __attribute__((ext_vector_type(2))) float v2f;
typedef __attribute__((ext_vector_type(4))) float v4f;
typedef __attribute__((ext_vector_type(8))) float v8f;

static __device__ __forceinline__ v8f wmma4(v2f a, v2f b, v8f c) {
    // D = A(16x4 f32) * B(4x16 f32) + C(16x16 f32)
    return __builtin_amdgcn_wmma_f32_16x16x4_f32(
        /*neg_a=*/false, a, /*neg_b=*/false, b,
        /*c_mod=*/(short)0, c, /*reuse_a=*/false, /*reuse_b=*/false);
}

static __device__ __forceinline__ v8f zero8() {
    v8f z;
#pragma unroll
    for (int i = 0; i < 8; ++i) z[i] = 0.0f;
    return z;
}

// ---------------------------------------------------------------------------
// Kernel 0: precompute sign(q_w) once (effective SignMaskLinear weight).
// Branch-free, vectorized by 4; removes all sign math from the GEMM hot loop.
// ---------------------------------------------------------------------------
__global__ __launch_bounds__(256)
void sign_kernel(const float* __restrict__ w, float* __restrict__ s)
{
    const int i = (blockIdx.x * 256 + threadIdx.x) * 4;
    v4f wv = *(const v4f*)(w + i);
    v4f sv;
#pragma unroll
    for (int c = 0; c < 4; ++c)
        sv[c] = (wv[c] > 0.f) ? 1.f : ((wv[c] < 0.f) ? -1.f : 0.f);
    *(v4f*)(s + i) = sv;
}

// ---------------------------------------------------------------------------
// Kernel 1: fused Q/K/V projection GEMM.  out = x @ W^T  (W row-major DxD).
// blockIdx.z selects {sign(q_w), k_w, v_w}; identical branch-free inner loop.
// 128 threads = 4 waves; each wave computes a 16x64 output tile.
// ---------------------------------------------------------------------------
__global__ __launch_bounds__(128)
void proj_gemm_kernel(const float* __restrict__ x,
                      const float* __restrict__ wq,
                      const float* __restrict__ wk,
                      const float* __restrict__ wv,
                      float* __restrict__ q, float* __restrict__ k,
                      float* __restrict__ v)
{
    const int lane = threadIdx.x & 31;
    const int wave = threadIdx.x >> 5;
    const int idx  = lane & 15;
    const int half = lane >> 4;
    const int mbase = (blockIdx.x * 4 + wave) * 16;
    const int nbase = blockIdx.y * 64;
    const int which = blockIdx.z;
    const float* w = (which == 0) ? wq : (which == 1 ? wk : wv);
    float* out     = (which == 0) ? q  : (which == 1 ? k  : v);

    const float* xrow = x + (size_t)(mbase + idx) * D_MODEL + 2 * half;
    const float* wrow0 = w + (size_t)(nbase + idx) * D_MODEL + 2 * half;

    v8f acc[4];
#pragma unroll
    for (int nt = 0; nt < 4; ++nt) acc[nt] = zero8();

#pragma unroll 4
    for (int kb = 0; kb < D_MODEL; kb += 4) {
        // A fragment: x[mbase+idx][kb + 2*half + {0,1}]   (8B-aligned)
        const v2f a = *(const v2f*)(xrow + kb);
#pragma unroll
        for (int nt = 0; nt < 4; ++nt) {
            // B fragment: B[kk][n] = W[nbase+nt*16+n][kk]
            const v2f b = *(const v2f*)(wrow0 + (size_t)nt * 16 * D_MODEL + kb);
            acc[nt] = wmma4(a, b, acc[nt]);
        }
    }
#pragma unroll
    for (int nt = 0; nt < 4; ++nt)
#pragma unroll
        for (int r = 0; r < 8; ++r)
            out[(size_t)(mbase + r + 8 * half) * D_MODEL + nbase + nt * 16 + idx] = acc[nt][r];
}

// ---------------------------------------------------------------------------
// Kernel 2: RMS-norm (+ per-head static top-k sparsify for q).
// 256 threads = 4 groups of 64; each group handles one (b,t,h) 64-vector.
// blockIdx.y: 0 -> q (norm + sparsify), 1 -> k (norm only).
// ---------------------------------------------------------------------------
__global__ __launch_bounds__(256)
void norm_kernel(float* __restrict__ q, float* __restrict__ k)
{
    __shared__ float sh[4][64];
    __shared__ float sth[4];
    const int g = threadIdx.x >> 6;
    const int t = threadIdx.x & 63;
    const bool is_q = (blockIdx.y == 0);
    float* buf = is_q ? q : k;
    const int vid = blockIdx.x * 4 + g;      // (b*T + tok)*16 + h
    const int h = vid & 15;

    float val = buf[(size_t)vid * 64 + t];
    sh[g][t] = val * val;
    __syncthreads();
#pragma unroll
    for (int off = 32; off; off >>= 1) {
        if (t < off) sh[g][t] += sh[g][t + off];
        __syncthreads();
    }
    const float rms = sqrtf(sh[g][0] * (1.0f / 64.0f)) + 1e-6f;
    float nv = val / rms;
    __syncthreads();

    if (is_q) {
        const float a = fabsf(nv);
        sh[g][t] = a;
        __syncthreads();
        // rank in descending order with index tie-break
        int rank = 0;
        for (int j = 0; j < 64; ++j) {
            float o = sh[g][j];
            rank += (o > a) || (o == a && j < t);
        }
        int keep = (int)((0.35f - 0.2f * (float)h / 15.0f) * 64.0f);
        if (keep < 1) keep = 1;
        if (rank == keep - 1) sth[g] = a;   // exactly one thread matches
        __syncthreads();
        const float th = sth[g];
        nv = (a >= th) ? nv : 0.0f;
        __syncthreads();
    }
    buf[(size_t)vid * 64 + t] = nv;
}

// ---------------------------------------------------------------------------
// Kernel 3: flash attention, one wave (32 threads) per (b,h,16-row q tile).
// Online softmax with running max/denominator and running sum(p^2) so that
// head_energy = mean(attn^2) is produced without materializing attn (saves a
// 256 MB intermediate; the whole working set then lives in the 192 MB L2).
// ---------------------------------------------------------------------------
__global__ __launch_bounds__(32)
void attn_kernel(const float* __restrict__ q, const float* __restrict__ k,
                 const float* __restrict__ v, float* __restrict__ att,
                 float* __restrict__ energy)
{
    __shared__ float pbuf[256];     // 16x16 P tile staging (C-layout -> A-layout)
    const int lane = threadIdx.x & 31;
    const int idx  = lane & 15;
    const int half = lane >> 4;
    const int qt = blockIdx.x;
    const int bh = blockIdx.y;
    const int b = bh >> 4, h = bh & 15;

    // Q tile A-fragments over the full Dh=64 (16 chunks of K=4)
    const float* qrow = q + (size_t)(b * T_ + qt * 16 + idx) * D_MODEL + h * HEAD_DIM + 2 * half;
    v2f aq[16];
#pragma unroll
    for (int c = 0; c < 16; ++c) aq[c] = *(const v2f*)(qrow + 4 * c);

    float m[8], l[8], s2[8];
#pragma unroll
    for (int r = 0; r < 8; ++r) { m[r] = -INFINITY; l[r] = 0.f; s2[r] = 0.f; }
    v8f O[4];
#pragma unroll
    for (int nt = 0; nt < 4; ++nt) O[nt] = zero8();

    for (int j = 0; j <= qt; ++j) {
        // S = q @ k^T  (16x16 tile)
        v8f S = zero8();
        const float* krow = k + (size_t)(b * T_ + j * 16 + idx) * D_MODEL + h * HEAD_DIM + 2 * half;
#pragma unroll
        for (int c = 0; c < 16; ++c) {
            const v2f bk = *(const v2f*)(krow + 4 * c);
            S = wmma4(aq[c], bk, S);
        }
        if (j == qt) {      // causal mask only touches the diagonal tile
#pragma unroll
            for (int r = 0; r < 8; ++r) {
                int rg = qt * 16 + r + 8 * half;
                int cg = j * 16 + idx;
                if (cg > rg) S[r] = -INFINITY;
            }
        }
        // online softmax per row (row r+8*half lives across lanes of one half)
        float alpha[8];
#pragma unroll
        for (int r = 0; r < 8; ++r) {
            float mx = S[r];
#pragma unroll
            for (int off = 8; off; off >>= 1) mx = fmaxf(mx, __shfl_xor(mx, off, 16));
            const float mn = fmaxf(m[r], mx);
            const float p  = __expf(S[r] - mn);
            float rs = p, rs2 = p * p;
#pragma unroll
            for (int off = 8; off; off >>= 1) {
                rs  += __shfl_xor(rs,  off, 16);
                rs2 += __shfl_xor(rs2, off, 16);
            }
            const float al = __expf(m[r] - mn);
            l[r]  = l[r]  * al + rs;
            s2[r] = s2[r] * al * al + rs2;
            m[r] = mn;
            alpha[r] = al;
            pbuf[(r + 8 * half) * 16 + idx] = p;
        }
#pragma unroll
        for (int nt = 0; nt < 4; ++nt)
#pragma unroll
            for (int r = 0; r < 8; ++r) O[nt][r] *= alpha[r];
        __syncthreads();
        // O += P @ V  (A = P from LDS, B = V tile)
#pragma unroll
        for (int c = 0; c < 4; ++c) {
            v2f pa;
            pa.x = pbuf[idx * 16 + 4 * c + 2 * half];
            pa.y = pbuf[idx * 16 + 4 * c + 2 * half + 1];
#pragma unroll
            for (int nt = 0; nt < 4; ++nt) {
                const size_t rv = (size_t)(b * T_ + j * 16 + 4 * c + 2 * half) * D_MODEL
                                  + h * HEAD_DIM + nt * 16 + idx;
                v2f bv; bv.x = v[rv]; bv.y = v[rv + D_MODEL];
                O[nt] = wmma4(pa, bv, O[nt]);
            }
        }
        __syncthreads();
    }

    // normalize + store head output (B,T,H*Dh) layout
#pragma unroll
    for (int r = 0; r < 8; ++r) {
        const float inv = 1.0f / l[r];
        const int rg = qt * 16 + r + 8 * half;
#pragma unroll
        for (int nt = 0; nt < 4; ++nt)
            att[(size_t)(b * T_ + rg) * D_MODEL + h * HEAD_DIM + nt * 16 + idx] = O[nt][r] * inv;
    }
    // head-energy partial: sum over this tile's 16 rows of sum(p^2)/Z^2
    if (idx == 0) {
        float e = 0.f;
#pragma unroll
        for (int r = 0; r < 8; ++r) e += s2[r] / (l[r] * l[r]);
        atomicAdd(&energy[b * N_HEAD + h], e * (1.0f / ((float)T_ * (float)T_)));
    }
}

// ---------------------------------------------------------------------------
// Kernel 4: entropy-driven head pruning -> 0/1 mask per (b,h).
// ---------------------------------------------------------------------------
__global__ void prune_kernel(const float* __restrict__ energy, float* __restrict__ mask)
{
    const int b = threadIdx.x;
    if (b >= B_) return;
    float e[N_HEAD];
    float mx = -INFINITY;
    for (int i = 0; i < N_HEAD; ++i) { e[i] = energy[b * N_HEAD + i]; mx = fmaxf(mx, e[i]); }
    float p[N_HEAD]; float Z = 0.f;
    for (int i = 0; i < N_HEAD; ++i) { p[i] = __expf(e[i] - mx); Z += p[i]; }
    float ent = 0.f;
    for (int i = 0; i < N_HEAD; ++i) { float pp = p[i] / Z; ent -= pp * logf(pp + 1e-9f); }
    float entn = ent / logf((float)N_HEAD);
    entn = fminf(fmaxf(entn, 0.f), 1.f);
    int keep = (int)rintf(2.0f + entn * 4.0f);   // MIN_KEEP_H + ent*(MAX-MIN), RNE
    if (keep < 1) keep = 1;
    if (keep > N_HEAD) keep = N_HEAD;
    float tmp[N_HEAD];
    for (int i = 0; i < N_HEAD; ++i) tmp[i] = e[i];
    for (int i = 0; i < N_HEAD; ++i) {           // selection sort, descending
        int best = i;
        for (int j = i + 1; j < N_HEAD; ++j) if (tmp[j] > tmp[best]) best = j;
        float t0 = tmp[i]; tmp[i] = tmp[best]; tmp[best] = t0;
    }
    const float th = tmp[keep - 1];
    for (int i = 0; i < N_HEAD; ++i)
        mask[b * N_HEAD + i] = (e[i] >= th) ? 1.0f : 0.0f;
}

// ---------------------------------------------------------------------------
// Kernel 5: output projection with head mask fused into the A loads.
// out = (att * mask) @ o_w^T      (mask^2 == mask, 0/1)
// ---------------------------------------------------------------------------
__global__ __launch_bounds__(128)
void out_gemm_kernel(const float* __restrict__ att, const float* __restrict__ ow,
                     const float* __restrict__ mask, float* __restrict__ out)
{
    const int lane = threadIdx.x & 31;
    const int wave = threadIdx.x >> 5;
    const int idx  = lane & 15;
    const int half = lane >> 4;
    const int mbase = (blockIdx.x * 4 + wave) * 16;
    const int nbase = blockIdx.y * 64;
    const int b = mbase >> 10;   // T == 1024, tiles never straddle batches

    const float* arow = att + (size_t)(mbase + idx) * D_MODEL + 2 * half;
    const float* wrow0 = ow + (size_t)(nbase + idx) * D_MODEL + 2 * half;

    v8f acc[4];
#pragma unroll
    for (int nt = 0; nt < 4; ++nt) acc[nt] = zero8();

#pragma unroll 4
    for (int kb = 0; kb < D_MODEL; kb += 4) {
        const float hm = mask[b * N_HEAD + (kb >> 6)];
        v2f a = *(const v2f*)(arow + kb);
        a.x *= hm; a.y *= hm;
#pragma unroll
        for (int nt = 0; nt < 4; ++nt) {
            const v2f bb = *(const v2f*)(wrow0 + (size_t)nt * 16 * D_MODEL + kb);
            acc[nt] = wmma4(a, bb, acc[nt]);
        }
    }
#pragma unroll
    for (int nt = 0; nt < 4; ++nt)
#pragma unroll
        for (int r = 0; r < 8; ++r)
            out[(size_t)(mbase + r + 8 * half) * D_MODEL + nbase + nt * 16 + idx] = acc[nt][r];
}

// ---------------------------------------------------------------------------
extern "C" void kernel_launch(void* const* d_in, const int* in_sizes, int n_in,
                              void* d_out, int out_size, void* d_ws, size_t ws_size,
                              hipStream_t stream)
{
    (void)in_sizes; (void)n_in; (void)out_size; (void)ws_size;
    // inputs: x, q_w, q_b, k_w, k_b, v_w, v_b, o_w, o_b (biases are zero)
    const float* x  = (const float*)d_in[0];
    const float* qw = (const float*)d_in[1];
    const float* kw = (const float*)d_in[3];
    const float* vw = (const float*)d_in[5];
    const float* ow = (const float*)d_in[7];
    float* out = (float*)d_out;

    const size_t SZ = (size_t)MROWS * D_MODEL;          // 4M floats
    const size_t WSZ = (size_t)D_MODEL * D_MODEL;       // 1M floats
    float* q      = (float*)d_ws;
    float* k      = q + SZ;
    float* v      = k + SZ;
    float* att    = v + SZ;
    float* qsign  = att + SZ;
    float* energy = qsign + WSZ;
    float* mask   = energy + B_ * N_HEAD;

    hipMemsetAsync(energy, 0, B_ * N_HEAD * sizeof(float), stream);

    sign_kernel<<<(D_MODEL * D_MODEL) / (256 * 4), 256, 0, stream>>>(qw, qsign);

    dim3 gproj(MROWS / 64, D_MODEL / 64, 3);
    proj_gemm_kernel<<<gproj, 128, 0, stream>>>(x, qsign, kw, vw, q, k, v);

    dim3 gnorm((B_ * T_ * N_HEAD) / 4, 2, 1);
    norm_kernel<<<gnorm, 256, 0, stream>>>(q, k);

    dim3 gattn(T_ / 16, B_ * N_HEAD, 1);
    attn_kernel<<<gattn, 32, 0, stream>>>(q, k, v, att, energy);

    prune_kernel<<<1, 32, 0, stream>>>(energy, mask);

    dim3 gout(MROWS / 64, D_MODEL / 64, 1);
    out_gemm_kernel<<<gout, 128, 0, stream>>>(att, ow, mask, out);
}